// GAT_70076686401563
// MI455X (gfx1250) — compile-verified
//
#include <hip/hip_runtime.h>

#define NN     100000
#define FIN    128
#define HIDC   32
#define HEADS  8
#define C1     256     // HEADS*HIDC
#define C2     32
#define NCLS   40
#define EDGES  1600000
#define ETOT   1700000 // EDGES + NN self loops
#define SLOPE  0.2f

typedef __attribute__((ext_vector_type(2))) float v2f;
typedef __attribute__((ext_vector_type(8))) float v8f;

__device__ __forceinline__ float lrelu(float x) { return x > 0.f ? x : SLOPE * x; }

// order-preserving float atomic max (works for mixed signs; init with -inf)
__device__ __forceinline__ void atomic_max_float(float* addr, float v) {
  if (v >= 0.f) atomicMax((int*)addr, __float_as_int(v));
  else          atomicMin((unsigned int*)addr, __float_as_uint(v));
}

// ---------------------------------------------------------------- fills
__global__ void fill_kernel(float* __restrict__ p, float v, long long n) {
  long long i = (long long)blockIdx.x * blockDim.x + threadIdx.x;
  if (i < n) p[i] = v;
}

// ------------------------------------------------- GEMM1: h1 = x @ W1
// x: [NN,128], W1: [128,256], out: [NN,256]. One WG (512 thr, 16 waves)
// handles 32 rows; wave w computes n-tile w for both 16-row m-tiles,
// sharing B fragments. W1 staged in LDS in K-pair-interleaved layout so a
// B fragment is one conflict-free ds_load_b64; x-block padded to 132.
#define WP1 544   // pair-row stride: 256*2 + 32  (mod 64 == 32 -> no conflicts)
#define XP1 132   // row stride: 128 + 4          (mod 64 == 4  -> no conflicts)
__global__ __launch_bounds__(512) void gemm1_kernel(
    const float* __restrict__ x, const float* __restrict__ W, float* __restrict__ out) {
  __shared__ float sW[(FIN / 2) * WP1]; // 64*544 floats = 136 KB
  __shared__ float sX[32 * XP1];        // 16.5 KB
  const int tid  = threadIdx.x;
  const int row0 = blockIdx.x * 32;

  for (int i = tid; i < (FIN * C1) / 4; i += 512) {       // 8192 float4
    float4 w = ((const float4*)W)[i];
    int k  = i >> 6;            // (i*4)/256
    int nb = (i & 63) << 2;     // (i*4)%256
    float* dst = &sW[(k >> 1) * WP1 + nb * 2 + (k & 1)];
    dst[0] = w.x; dst[2] = w.y; dst[4] = w.z; dst[6] = w.w;
  }
  for (int i = tid; i < (32 * FIN) / 4; i += 512) {       // 1024 float4
    int r = i >> 5;             // FIN/4 = 32 per row
    int c = (i & 31) << 2;
    float4 v = ((const float4*)(x + (size_t)row0 * FIN))[i];
    *(float4*)&sX[r * XP1 + c] = v;
  }
  __syncthreads();

  const int wave = tid >> 5, lane = tid & 31;
  const int l = lane & 15, hi = lane >> 4;
  const int nt = wave;                      // 16 waves -> 16 n-tiles
  v8f acc0 = {}, acc1 = {};
  const float* ax0 = &sX[(l)      * XP1 + 2 * hi];
  const float* ax1 = &sX[(16 + l) * XP1 + 2 * hi];
  const float* bw  = &sW[hi * WP1 + (nt * 16 + l) * 2];
  #pragma unroll 4
  for (int k0 = 0; k0 < FIN; k0 += 4) {
    const float* bp = bw + (k0 >> 1) * WP1;
    v2f b; b.x = bp[0]; b.y = bp[1];
    v2f a0; a0.x = ax0[k0]; a0.y = ax0[k0 + 1];
    v2f a1; a1.x = ax1[k0]; a1.y = ax1[k0 + 1];
    acc0 = __builtin_amdgcn_wmma_f32_16x16x4_f32(false, a0, false, b, (short)0, acc0, false, false);
    acc1 = __builtin_amdgcn_wmma_f32_16x16x4_f32(false, a1, false, b, (short)0, acc1, false, false);
  }
  float* o0 = out + (size_t)(row0 +      8 * hi) * C1 + nt * 16 + l;
  float* o1 = out + (size_t)(row0 + 16 + 8 * hi) * C1 + nt * 16 + l;
  #pragma unroll
  for (int i = 0; i < 8; ++i) { o0[(size_t)i * C1] = acc0[i]; o1[(size_t)i * C1] = acc1[i]; }
}

// ------------------------------------------------- GEMM2: h2 = elu1 @ W2
// in: [NN,256], W2: [256,32], out: [NN,32]. WG = 256 thr (8 waves) covers
// 64 rows -> 4 m-tiles x 2 n-tiles = 8 tiles, one per wave.
#define WP2 96    // pair-row stride: 32*2 + 32 (mod 64 == 32)
#define XP2 260   // row stride: 256 + 4       (mod 64 == 4)
__global__ __launch_bounds__(256) void gemm2_kernel(
    const float* __restrict__ xin, const float* __restrict__ W, float* __restrict__ out) {
  __shared__ float sW[(C1 / 2) * WP2];  // 48 KB
  __shared__ float sX[64 * XP2];        // 65 KB
  const int tid  = threadIdx.x;
  const int row0 = blockIdx.x * 64;

  for (int i = tid; i < (C1 * C2) / 4; i += 256) {        // 2048 float4
    float4 w = ((const float4*)W)[i];
    int k  = i >> 3;            // (i*4)/32
    int nb = (i & 7) << 2;
    float* dst = &sW[(k >> 1) * WP2 + nb * 2 + (k & 1)];
    dst[0] = w.x; dst[2] = w.y; dst[4] = w.z; dst[6] = w.w;
  }
  for (int i = tid; i < (64 * C1) / 4; i += 256) {        // 4096 float4
    int r = i >> 6;             // C1/4 = 64 per row
    int c = (i & 63) << 2;
    float4 v = {0.f, 0.f, 0.f, 0.f};
    if (row0 + r < NN) v = *(const float4*)(xin + (size_t)(row0 + r) * C1 + c);
    *(float4*)&sX[r * XP2 + c] = v;
  }
  __syncthreads();

  const int wave = tid >> 5, lane = tid & 31;
  const int l = lane & 15, hi = lane >> 4;
  const int mt = wave >> 1, nt = wave & 1;
  v8f acc = {};
  const float* ax = &sX[(mt * 16 + l) * XP2 + 2 * hi];
  const float* bw = &sW[hi * WP2 + (nt * 16 + l) * 2];
  #pragma unroll 4
  for (int k0 = 0; k0 < C1; k0 += 4) {
    const float* bp = bw + (k0 >> 1) * WP2;
    v2f b; b.x = bp[0]; b.y = bp[1];
    v2f a; a.x = ax[k0]; a.y = ax[k0 + 1];
    acc = __builtin_amdgcn_wmma_f32_16x16x4_f32(false, a, false, b, (short)0, acc, false, false);
  }
  const int mbase = row0 + mt * 16 + 8 * hi;
  float* orow = out + (size_t)mbase * C2 + nt * 16 + l;
  #pragma unroll
  for (int i = 0; i < 8; ++i)
    if (mbase + i < NN) orow[(size_t)i * C2] = acc[i];
}

// ----------------------------------------- per-(node,head) attention logits
template <int H>
__global__ void att_kernel(const float* __restrict__ h, const float* __restrict__ atts,
                           const float* __restrict__ attd, float* __restrict__ as_,
                           float* __restrict__ ad_) {
  int id = blockIdx.x * 256 + threadIdx.x;   // id = node*H + head
  if (id >= NN * H) return;
  int head = id & (H - 1);
  const float4* hv = (const float4*)(h + (size_t)id * HIDC);
  const float4* sv = (const float4*)(atts + head * HIDC);
  const float4* dv = (const float4*)(attd + head * HIDC);
  float s = 0.f, d = 0.f;
  #pragma unroll
  for (int i = 0; i < HIDC / 4; ++i) {
    float4 a = hv[i], b = sv[i], c = dv[i];
    s += a.x * b.x + a.y * b.y + a.z * b.z + a.w * b.w;
    d += a.x * c.x + a.y * c.y + a.z * c.z + a.w * c.w;
  }
  as_[id] = s; ad_[id] = d;
}

// ----------------------------------------- edge passes (segment softmax)
template <int H>
__global__ void edge_max_kernel(const int* __restrict__ ei, const float* __restrict__ as_,
                                const float* __restrict__ ad_, float* __restrict__ emax) {
  int e = blockIdx.x * 256 + threadIdx.x;
  if (e >= ETOT) return;
  int s, d;
  if (e < EDGES) { s = ei[e]; d = ei[EDGES + e]; } else { s = e - EDGES; d = s; }
  #pragma unroll
  for (int h = 0; h < H; ++h)
    atomic_max_float(&emax[(size_t)d * H + h],
                     lrelu(as_[(size_t)s * H + h] + ad_[(size_t)d * H + h]));
}

template <int H>
__global__ void edge_expsum_kernel(const int* __restrict__ ei, const float* __restrict__ as_,
                                   const float* __restrict__ ad_, const float* __restrict__ emax,
                                   float* __restrict__ ee, float* __restrict__ den) {
  int e = blockIdx.x * 256 + threadIdx.x;
  if (e >= ETOT) return;
  int s, d;
  if (e < EDGES) { s = ei[e]; d = ei[EDGES + e]; } else { s = e - EDGES; d = s; }
  #pragma unroll
  for (int h = 0; h < H; ++h) {
    float v  = lrelu(as_[(size_t)s * H + h] + ad_[(size_t)d * H + h]);
    float ex = __expf(v - emax[(size_t)d * H + h]);
    ee[(size_t)e * H + h] = ex;
    atomicAdd(&den[(size_t)d * H + h], ex);
  }
}

template <int COLS, int H>
__global__ void edge_agg_kernel(const int* __restrict__ ei, const float* __restrict__ feat,
                                const float* __restrict__ ee, const float* __restrict__ den,
                                float* __restrict__ out) {
  constexpr int EPB = 256 / COLS;
  int e = blockIdx.x * EPB + threadIdx.x / COLS;
  if (e >= ETOT) return;
  int col = threadIdx.x & (COLS - 1);
  int s, d;
  if (e < EDGES) { s = ei[e]; d = ei[EDGES + e]; } else { s = e - EDGES; d = s; }
  int h = col >> 5;   // 32 channels per head
  float alpha = ee[(size_t)e * H + h] / den[(size_t)d * H + h];
  atomicAdd(&out[(size_t)d * COLS + col], alpha * feat[(size_t)s * COLS + col]);
}

// ----------------------------------------- bias + ELU (cols is a power of 2)
__global__ void bias_elu_kernel(float* __restrict__ v, const float* __restrict__ b,
                                int colmask, long long n) {
  long long i = (long long)blockIdx.x * 256 + threadIdx.x;
  if (i >= n) return;
  float x = v[i] + b[(int)(i & colmask)];
  v[i] = x > 0.f ? x : (__expf(x) - 1.f);
}

// ----------------------------------------- bias2 + ELU + classifier + log_softmax
__global__ __launch_bounds__(256) void final_kernel(
    const float* __restrict__ out2, const float* __restrict__ b2,
    const float* __restrict__ Wout, const float* __restrict__ bout,
    float* __restrict__ y) {
  __shared__ float sW[HIDC * NCLS];
  __shared__ float sb[NCLS];
  __shared__ float sb2[HIDC];
  for (int i = threadIdx.x; i < HIDC * NCLS; i += 256) sW[i] = Wout[i];
  if (threadIdx.x < NCLS) sb[threadIdx.x] = bout[threadIdx.x];
  if (threadIdx.x < HIDC) sb2[threadIdx.x] = b2[threadIdx.x];
  __syncthreads();
  int n = blockIdx.x * 256 + threadIdx.x;
  if (n >= NN) return;
  float hv[HIDC];
  #pragma unroll
  for (int c = 0; c < HIDC; ++c) {
    float xx = out2[(size_t)n * HIDC + c] + sb2[c];
    hv[c] = xx > 0.f ? xx : (__expf(xx) - 1.f);
  }
  float lg[NCLS];
  float mx = -1e30f;
  #pragma unroll
  for (int k = 0; k < NCLS; ++k) {
    float acc = sb[k];
    #pragma unroll
    for (int c = 0; c < HIDC; ++c) acc += hv[c] * sW[c * NCLS + k];
    lg[k] = acc;
    mx = fmaxf(mx, acc);
  }
  float sum = 0.f;
  #pragma unroll
  for (int k = 0; k < NCLS; ++k) sum += __expf(lg[k] - mx);
  float lse = __logf(sum);
  #pragma unroll
  for (int k = 0; k < NCLS; ++k) y[(size_t)n * NCLS + k] = lg[k] - mx - lse;
}

// ---------------------------------------------------------------- launch
extern "C" void kernel_launch(void* const* d_in, const int* in_sizes, int n_in,
                              void* d_out, int out_size, void* d_ws, size_t ws_size,
                              hipStream_t stream) {
  (void)in_sizes; (void)n_in; (void)out_size; (void)ws_size;
  const float* x      = (const float*)d_in[0];
  const int*   ei     = (const int*)d_in[1];
  const float* W1     = (const float*)d_in[2];
  const float* att_s1 = (const float*)d_in[3];
  const float* att_d1 = (const float*)d_in[4];
  const float* b1     = (const float*)d_in[5];
  const float* W2     = (const float*)d_in[6];
  const float* att_s2 = (const float*)d_in[7];
  const float* att_d2 = (const float*)d_in[8];
  const float* b2     = (const float*)d_in[9];
  const float* Wout   = (const float*)d_in[10];
  const float* bout   = (const float*)d_in[11];
  float* ws = (float*)d_ws;

  // workspace layout (floats); layer-2 buffers alias the dead h1 region
  float* h1    = ws;                          // NN*256
  float* out1  = h1    + (size_t)NN * C1;     // NN*256
  float* as1   = out1  + (size_t)NN * C1;     // NN*8
  float* ad1   = as1   + (size_t)NN * HEADS;
  float* emax1 = ad1   + (size_t)NN * HEADS;
  float* den1  = emax1 + (size_t)NN * HEADS;
  float* ee1   = den1  + (size_t)NN * HEADS;  // ETOT*8
  float* h2    = h1;                          // NN*32 (h1 dead after layer-1 agg)
  float* as2   = h2    + (size_t)NN * C2;
  float* ad2   = as2   + NN;
  float* emax2 = ad2   + NN;
  float* den2  = emax2 + NN;
  float* out2  = den2  + NN;                  // NN*32
  float* ee2   = out2  + (size_t)NN * C2;     // ETOT

  const float NEG_INF = -__builtin_inff();

  // ---- layer 1 ----
  gemm1_kernel<<<NN / 32, 512, 0, stream>>>(x, W1, h1);
  att_kernel<HEADS><<<(NN * HEADS + 255) / 256, 256, 0, stream>>>(h1, att_s1, att_d1, as1, ad1);
  fill_kernel<<<(NN * HEADS + 255) / 256, 256, 0, stream>>>(emax1, NEG_INF, (long long)NN * HEADS);
  fill_kernel<<<(NN * HEADS + 255) / 256, 256, 0, stream>>>(den1, 0.f, (long long)NN * HEADS);
  fill_kernel<<<(unsigned)(((size_t)NN * C1 + 255) / 256), 256, 0, stream>>>(out1, 0.f, (long long)NN * C1);
  edge_max_kernel<HEADS><<<(ETOT + 255) / 256, 256, 0, stream>>>(ei, as1, ad1, emax1);
  edge_expsum_kernel<HEADS><<<(ETOT + 255) / 256, 256, 0, stream>>>(ei, as1, ad1, emax1, ee1, den1);
  edge_agg_kernel<C1, HEADS><<<ETOT, 256, 0, stream>>>(ei, h1, ee1, den1, out1);
  bias_elu_kernel<<<(unsigned)(((size_t)NN * C1 + 255) / 256), 256, 0, stream>>>(out1, b1, C1 - 1, (long long)NN * C1);

  // ---- layer 2 ----
  gemm2_kernel<<<(NN + 63) / 64, 256, 0, stream>>>(out1, W2, h2);
  att_kernel<1><<<(NN + 255) / 256, 256, 0, stream>>>(h2, att_s2, att_d2, as2, ad2);
  fill_kernel<<<(NN + 255) / 256, 256, 0, stream>>>(emax2, NEG_INF, NN);
  fill_kernel<<<(NN + 255) / 256, 256, 0, stream>>>(den2, 0.f, NN);
  fill_kernel<<<(unsigned)(((size_t)NN * C2 + 255) / 256), 256, 0, stream>>>(out2, 0.f, (long long)NN * C2);
  edge_max_kernel<1><<<(ETOT + 255) / 256, 256, 0, stream>>>(ei, as2, ad2, emax2);
  edge_expsum_kernel<1><<<(ETOT + 255) / 256, 256, 0, stream>>>(ei, as2, ad2, emax2, ee2, den2);
  edge_agg_kernel<C2, 1><<<(ETOT + 7) / 8, 256, 0, stream>>>(ei, h2, ee2, den2, out2);

  // ---- classifier + log_softmax ----
  final_kernel<<<(NN + 255) / 256, 256, 0, stream>>>(out2, b2, Wout, bout, (float*)d_out);
}